// PitcherBatterGNN_24696061952120
// MI455X (gfx1250) — compile-verified
//
#include <hip/hip_runtime.h>

typedef float v2f __attribute__((ext_vector_type(2)));
typedef float v8f __attribute__((ext_vector_type(8)));

#define CH 64  // IN_CH == HID == 64

// ---------------------------------------------------------------- utilities
__global__ void zero_f32_kernel(float* __restrict__ p, long long n4) {
  long long i = (long long)blockIdx.x * blockDim.x + threadIdx.x;
  if (i < n4) reinterpret_cast<float4*>(p)[i] = make_float4(0.f, 0.f, 0.f, 0.f);
}

__global__ void degree_kernel(const int* __restrict__ dst, float* __restrict__ cnt, int E) {
  int e = blockIdx.x * blockDim.x + threadIdx.x;
  if (e < E) atomicAdd(&cnt[dst[e]], 1.0f);
}

// 64 threads per edge: coalesced feature read, f32 atomic scatter (lands in 192MB L2)
__global__ void scatter_add_kernel(const float* __restrict__ feat, const int* __restrict__ src,
                                   const int* __restrict__ dst, float* __restrict__ agg, int E) {
  long long t = (long long)blockIdx.x * blockDim.x + threadIdx.x;
  int e = (int)(t >> 6);
  if (e >= E) return;
  int c = (int)(t & 63);
  int s = src[e], d = dst[e];
  atomicAdd(agg + (size_t)d * CH + c, feat[(size_t)s * CH + c]);
}

// ------------------------------------------------- fused SAGE layer (WMMA f32)
// out[n,:] = relu( (agg[n]/max(cnt,1)) @ Wl + bl + self[n] @ Wr )
// block = 128 thr = 4 wave32; each wave owns one 16x16 f32 tile (cols = wave*16..+15)
__global__ __launch_bounds__(128) void sage_layer_kernel(
    const float* __restrict__ self_feat, const float* __restrict__ agg,
    const float* __restrict__ cnt, const float* __restrict__ Wl,
    const float* __restrict__ bl, const float* __restrict__ Wr,
    float* __restrict__ out, int N) {
  const int lane = threadIdx.x & 31;
  const int wave = threadIdx.x >> 5;
  const int lrow = lane & 15;
  const int hi   = lane >> 4;
  const int row_base = blockIdx.x * 16;
  const int col = wave * 16 + lrow;

  int row  = row_base + lrow;
  int arow = row < N ? row : N - 1;               // clamp (no EXEC divergence)
  float inv = 1.0f / fmaxf(cnt[arow], 1.0f);
  const float* aggr  = agg       + (size_t)arow * CH;
  const float* selfr = self_feat + (size_t)arow * CH;

  v8f acc = {};
  #pragma unroll
  for (int s = 0; s < 16; ++s) {                  // mean @ Wl
    int k = 4 * s + 2 * hi;
    float2 av = *reinterpret_cast<const float2*>(aggr + k);
    v2f a; a.x = av.x * inv; a.y = av.y * inv;
    v2f b; b.x = Wl[k * CH + col]; b.y = Wl[(k + 1) * CH + col];
    acc = __builtin_amdgcn_wmma_f32_16x16x4_f32(false, a, false, b, (short)0, acc, false, false);
  }
  #pragma unroll
  for (int s = 0; s < 16; ++s) {                  // self @ Wr (same accumulator)
    int k = 4 * s + 2 * hi;
    float2 av = *reinterpret_cast<const float2*>(selfr + k);
    v2f a; a.x = av.x; a.y = av.y;
    v2f b; b.x = Wr[k * CH + col]; b.y = Wr[(k + 1) * CH + col];
    acc = __builtin_amdgcn_wmma_f32_16x16x4_f32(false, a, false, b, (short)0, acc, false, false);
  }

  float bias = bl[col];
  #pragma unroll
  for (int r = 0; r < 8; ++r) {                   // C layout: row = r + 8*hi, col = lane&15
    int orow = row_base + r + 8 * hi;
    if (orow < N) out[(size_t)orow * CH + col] = fmaxf(acc[r] + bias, 0.0f);
  }
}

// --------------------------------------------- fused pair MLP (WMMA + reduce)
// out[p] = relu(concat(h2[i0],h2[i1]) @ Wr1 + br1) @ Wr2 + br2 ; hidden stays in regs
__global__ __launch_bounds__(128) void pair_mlp_kernel(
    const float* __restrict__ h2, const int* __restrict__ pairs,
    const float* __restrict__ Wr1, const float* __restrict__ br1,
    const float* __restrict__ Wr2, const float* __restrict__ br2,
    float* __restrict__ out, int P) {
  __shared__ float red[4][16];
  const int lane = threadIdx.x & 31;
  const int wave = threadIdx.x >> 5;
  const int lrow = lane & 15;
  const int hi   = lane >> 4;
  const int pair_base = blockIdx.x * 16;
  const int col = wave * 16 + lrow;

  int p  = pair_base + lrow;
  int ap = p < P ? p : P - 1;
  int i0 = pairs[2 * ap], i1 = pairs[2 * ap + 1];
  const float* r0 = h2 + (size_t)i0 * CH;
  const float* r1 = h2 + (size_t)i1 * CH;

  v8f acc = {};
  #pragma unroll
  for (int s = 0; s < 32; ++s) {                  // K = 128 (two concatenated halves)
    int k = 4 * s + 2 * hi;
    const float* srcp = (k < CH) ? (r0 + k) : (r1 + (k - CH));   // select, not branch
    float2 av = *reinterpret_cast<const float2*>(srcp);
    v2f a; a.x = av.x; a.y = av.y;
    v2f b; b.x = Wr1[k * 64 + col]; b.y = Wr1[(k + 1) * 64 + col];
    acc = __builtin_amdgcn_wmma_f32_16x16x4_f32(false, a, false, b, (short)0, acc, false, false);
  }

  float bias = br1[col];
  float w2   = Wr2[col];
  float part[8];
  #pragma unroll
  for (int r = 0; r < 8; ++r) {                   // relu, *Wr2, reduce 16 cols in-wave
    float v = fmaxf(acc[r] + bias, 0.0f) * w2;
    v += __shfl_xor(v, 1, 32);
    v += __shfl_xor(v, 2, 32);
    v += __shfl_xor(v, 4, 32);
    v += __shfl_xor(v, 8, 32);
    part[r] = v;
  }
  if (lrow == 0) {
    #pragma unroll
    for (int r = 0; r < 8; ++r) red[wave][r + 8 * hi] = part[r];
  }
  __syncthreads();
  if (threadIdx.x < 16) {                         // cross-wave reduce (4 col-groups)
    int op = pair_base + threadIdx.x;
    if (op < P)
      out[op] = red[0][threadIdx.x] + red[1][threadIdx.x] +
                red[2][threadIdx.x] + red[3][threadIdx.x] + br2[0];
  }
}

// -------------------------------------------------------------------- driver
extern "C" void kernel_launch(void* const* d_in, const int* in_sizes, int n_in,
                              void* d_out, int out_size, void* d_ws, size_t ws_size,
                              hipStream_t stream) {
  (void)n_in; (void)out_size; (void)ws_size;
  const float* x   = (const float*)d_in[0];
  const int*   ei  = (const int*)d_in[1];
  const int*   prs = (const int*)d_in[2];
  const float* W1l = (const float*)d_in[3];
  const float* b1l = (const float*)d_in[4];
  const float* W1r = (const float*)d_in[5];
  const float* W2l = (const float*)d_in[6];
  const float* b2l = (const float*)d_in[7];
  const float* W2r = (const float*)d_in[8];
  const float* Wr1 = (const float*)d_in[9];
  const float* br1 = (const float*)d_in[10];
  const float* Wr2 = (const float*)d_in[11];
  const float* br2 = (const float*)d_in[12];

  const int N = in_sizes[0] / CH;     // 100000
  const int E = in_sizes[1] / 2;      // 1280000
  const int P = in_sizes[2] / 2;      // 500000
  const int* src = ei;
  const int* dst = ei + E;

  float* ws  = (float*)d_ws;          // cnt | agg | h1 | h2  (~77 MB total)
  float* cnt = ws;
  float* agg = ws + N;
  float* h1  = agg + (size_t)N * CH;
  float* h2  = h1  + (size_t)N * CH;

  const int ZB = 256;
  long long z1 = ((long long)N + (long long)N * CH) / 4;   // cnt+agg contiguous
  long long z2 = ((long long)N * CH) / 4;
  long long st = (long long)E * 64;
  int sgrid = (int)((st + 255) / 256);
  int ggrid = (N + 15) / 16;

  // Layer 1
  zero_f32_kernel<<<(int)((z1 + ZB - 1) / ZB), ZB, 0, stream>>>(cnt, z1);
  degree_kernel<<<(E + 255) / 256, 256, 0, stream>>>(dst, cnt, E);
  scatter_add_kernel<<<sgrid, 256, 0, stream>>>(x, src, dst, agg, E);
  sage_layer_kernel<<<ggrid, 128, 0, stream>>>(x, agg, cnt, W1l, b1l, W1r, h1, N);
  // Layer 2 (degrees unchanged; re-zero + re-scatter agg)
  zero_f32_kernel<<<(int)((z2 + ZB - 1) / ZB), ZB, 0, stream>>>(agg, z2);
  scatter_add_kernel<<<sgrid, 256, 0, stream>>>(h1, src, dst, agg, E);
  sage_layer_kernel<<<ggrid, 128, 0, stream>>>(h1, agg, cnt, W2l, b2l, W2r, h2, N);
  // Pair MLP
  pair_mlp_kernel<<<(P + 15) / 16, 128, 0, stream>>>(h2, prs, Wr1, br1, Wr2, br2,
                                                     (float*)d_out, P);
}